// FrontBone_87969520157428
// MI455X (gfx1250) — compile-verified
//
#include <hip/hip_runtime.h>
#include <math.h>

typedef float v2f __attribute__((ext_vector_type(2)));
typedef float v8f __attribute__((ext_vector_type(8)));

#define BATCH   16
#define CHN     256
#define DD      256
#define HW      1024
#define TILE_N  16

// ---------------------------------------------------------------------------
// Kernel 0: per batch b, t[c] = one_hot[b]·Wt[c,:] + bt[c];
//           sk1[h1[c]] += t[c]*s1[c]  (count sketch, spatially constant)
// ---------------------------------------------------------------------------
__global__ void sk1_kernel(const float* __restrict__ one_hot,
                           const float* __restrict__ Wt,
                           const float* __restrict__ bt,
                           const float* __restrict__ s1,
                           const int*   __restrict__ h1,
                           float* __restrict__ sk1_out) {
  __shared__ float sk1[DD];
  const int b = blockIdx.x;
  const int c = threadIdx.x;            // 256 threads
  sk1[c] = 0.0f;
  __syncthreads();
  const float4* oh4 = (const float4*)(one_hot + b * 1000);
  const float4* w4  = (const float4*)(Wt + c * 1000);
  float t = bt[c];
  for (int j = 0; j < 250; ++j) {       // 1000 = 250 * 4
    const float4 a = oh4[j], w = w4[j];
    t += a.x * w.x + a.y * w.y + a.z * w.z + a.w * w.w;
  }
  atomicAdd(&sk1[h1[c]], t * s1[c]);    // ds_add_f32
  __syncthreads();
  sk1_out[b * DD + c] = sk1[c];
}

// ---------------------------------------------------------------------------
// Kernel 1: fused circulant-GEMM (WMMA fp32) + WMMA W1/relu + sigmoid + att*P5
// One block = one batch x one tile of 16 spatial positions. 128 thr = 4 waves.
// ---------------------------------------------------------------------------
__global__ void mcb_att_kernel(const float* __restrict__ P5,
                               const float* __restrict__ W1,   // [32,256]
                               const float* __restrict__ W2,   // [1,32]
                               const float* __restrict__ s2,
                               const int*   __restrict__ h2,
                               const float* __restrict__ sk1g, // [B,256]
                               float* __restrict__ out_att,    // [B,1,32,32]
                               float* __restrict__ out_feat)   // [B,256,32,32]
{
  __shared__ __align__(16) float p5t [CHN * TILE_N];  // 16 KB  P5 tile  [c][n]
  __shared__ __align__(16) float sk2 [DD  * TILE_N];  // 16 KB  sketch2  [k][n]
  __shared__ __align__(16) float mcbt[DD  * TILE_N];  // 16 KB  conv out [d][n]
  __shared__ __align__(16) float rsk1[2 * DD + 8];    //  2 KB  reversed sk1
  __shared__ __align__(16) float hmid[32 * TILE_N];   //  2 KB
  __shared__ __align__(16) float attn[TILE_N];

  const int ct   = blockIdx.x;           // 0..63 spatial tile
  const int b    = blockIdx.y;           // 0..15 batch
  const int t    = threadIdx.x;          // 0..127
  const int lane = t & 31;
  const int wave = t >> 5;

  // ---- load P5 tile (b128), zero sk2 (b128), load reversed+doubled sk1 ----
  const float* p5g = P5 + (size_t)b * CHN * HW + (size_t)ct * TILE_N;
  {
    const int q = t & 3;                 // quad of 4 floats within a row
    for (int c = (t >> 2); c < CHN; c += 32)
      *(float4*)&p5t[c * TILE_N + q * 4] =
          *(const float4*)(p5g + (size_t)c * HW + q * 4);
    const float4 z4 = {0.f, 0.f, 0.f, 0.f};
    for (int i = t; i < DD * TILE_N / 4; i += 128) ((float4*)sk2)[i] = z4;
    // rsk1[i] = sk1[(-i) mod 256], i in [0, 520): ascending-index circulant
    for (int i = t; i < 2 * DD + 8; i += 128)
      rsk1[i] = sk1g[b * DD + ((-i) & 255)];
  }
  __syncthreads();

  // ---- count-sketch scatter of P5 tile into sk2 (LDS atomics) ----
  for (int i = t; i < CHN * TILE_N; i += 128) {
    const int c = i >> 4, n = i & 15;
    atomicAdd(&sk2[h2[c] * TILE_N + n], p5t[i] * s2[c]);
  }
  __syncthreads();

  // ---- circulant GEMM: mcb[d][n] = sum_k sk1[(d-k)&255] * sk2[k][n] ----
  // fp32 16x16x4 layouts:
  //   A: lane m=lane&15, VGPR v holds K = v + 2*(lane>>4)
  //   B: lane n=lane&15, VGPR v holds K = v + 2*(lane>>4)
  //   C/D: VGPR v holds M = v + 8*(lane>>4), N = lane&15
  const int n     = lane & 15;
  const int khalf = (lane >> 4) << 1;    // 0 or 2
  for (int mt4 = 0; mt4 < 4; ++mt4) {
    const int mt = wave * 4 + mt4;
    v8f acc = {};
    // A[m][kG]   = sk1[(d-kG)&255]   = rsk1[ja]
    // A[m][kG+1] = sk1[(d-kG-1)&255] = rsk1[ja+1],  ja = kG - d + 256, +4/step
    int ja = khalf - (mt * 16 + n) + 256;
    const float* bp = &sk2[khalf * TILE_N + n];  // B base, +64 floats per step
#pragma unroll 8
    for (int kk = 0; kk < 64; ++kk) {
      v2f a, bf;
      a.x  = rsk1[ja];
      a.y  = rsk1[ja + 1];
      bf.x = bp[0];
      bf.y = bp[TILE_N];
      acc = __builtin_amdgcn_wmma_f32_16x16x4_f32(
          /*neg_a=*/false, a, /*neg_b=*/false, bf,
          /*c_mod=*/(short)0, acc, /*reuse_a=*/false, /*reuse_b=*/false);
      ja += 4;
      bp += 4 * TILE_N;
    }
    const int mbase = mt * 16 + ((lane >> 4) << 3);
#pragma unroll
    for (int v = 0; v < 8; ++v)
      mcbt[(mbase + v) * TILE_N + n] = acc[v];
  }
  __syncthreads();

  // ---- hmid = relu(W1 @ mcb) as WMMA: waves 0,1 -> one 16x16 tile each ----
  if (wave < 2) {
    const int mt = wave;                 // output rows mt*16 .. mt*16+15
    v8f acc = {};
    const float* wp = W1 + (size_t)(mt * 16 + n) * DD + khalf; // b64 per step
    const float* bp = &mcbt[khalf * TILE_N + n];
#pragma unroll 8
    for (int kk = 0; kk < 64; ++kk) {
      const v2f a = *(const v2f*)wp;     // W1[row][kG], W1[row][kG+1]
      v2f bf;
      bf.x = bp[0];
      bf.y = bp[TILE_N];
      acc = __builtin_amdgcn_wmma_f32_16x16x4_f32(
          false, a, false, bf, (short)0, acc, false, false);
      wp += 4;
      bp += 4 * TILE_N;
    }
    const int obase = mt * 16 + ((lane >> 4) << 3);
#pragma unroll
    for (int v = 0; v < 8; ++v)
      hmid[(obase + v) * TILE_N + n] = fmaxf(acc[v], 0.f);
  }
  __syncthreads();

  // ---- att = sigmoid(W2 @ hmid) ----
  if (t < TILE_N) {
    float s = 0.f;
    for (int o = 0; o < 32; ++o) s += hmid[o * TILE_N + t] * W2[o];
    const float a = 1.0f / (1.0f + expf(-s));
    attn[t] = a;
    out_att[b * HW + ct * TILE_N + t] = a;
  }
  __syncthreads();

  // ---- att_feat = att * P5 (reuse resident P5 tile, b128 stores) ----
  {
    float* outf = out_feat + (size_t)b * CHN * HW + (size_t)ct * TILE_N;
    const int q = t & 3;
    const float4 a4 = *(const float4*)&attn[q * 4];
    for (int c = (t >> 2); c < CHN; c += 32) {
      float4 v = *(const float4*)&p5t[c * TILE_N + q * 4];
      v.x *= a4.x; v.y *= a4.y; v.z *= a4.z; v.w *= a4.w;
      *(float4*)(outf + (size_t)c * HW + q * 4) = v;
    }
  }
}

// ---------------------------------------------------------------------------
extern "C" void kernel_launch(void* const* d_in, const int* in_sizes, int n_in,
                              void* d_out, int out_size, void* d_ws, size_t ws_size,
                              hipStream_t stream) {
  // setup_inputs order: P3, P4, P5, one_hot, Wt, bt, W1, W2, s1, s2, h1, h2
  const float* P5      = (const float*)d_in[2];
  const float* one_hot = (const float*)d_in[3];
  const float* Wt      = (const float*)d_in[4];
  const float* bt      = (const float*)d_in[5];
  const float* W1      = (const float*)d_in[6];
  const float* W2      = (const float*)d_in[7];
  const float* s1      = (const float*)d_in[8];
  const float* s2      = (const float*)d_in[9];
  const int*   h1      = (const int*)d_in[10];
  const int*   h2      = (const int*)d_in[11];

  float* out     = (float*)d_out;                 // [0,16384): att_map5
  float* outfeat = out + BATCH * HW;              // then att_feat5
  float* sk1_ws  = (float*)d_ws;                  // [16,256] scratch

  sk1_kernel<<<dim3(BATCH), dim3(256), 0, stream>>>(one_hot, Wt, bt, s1, h1, sk1_ws);
  mcb_att_kernel<<<dim3(HW / TILE_N, BATCH), dim3(128), 0, stream>>>(
      P5, W1, W2, s2, h2, sk1_ws, out, outfeat);
}